// CascadingSinkCacheTriton_84817014161727
// MI455X (gfx1250) — compile-verified
//
#include <hip/hip_runtime.h>

// ---- problem constants (mirror reference) ----
#define NUM_SINK   4
#define WINDOW     512
#define CASCADES   4
#define HID        128
#define K_TOK      4096
#define T_SLOTS    (CASCADES * WINDOW)   // 2048 cache slots per sequence
#define MAX_SEQ    T_SLOTS               // warm-up horizon
#define ROW        (2 * HID)             // 256 floats per output slot (k|v)
#define PDEPTH     8                     // async staging ring depth (tokens)

#define AS_GLOBAL __attribute__((address_space(1)))
#define AS_LDS    __attribute__((address_space(3)))

// 16-byte vector type matching the builtin's expected pointee
typedef int v4i __attribute__((vector_size(4 * sizeof(int))));

// ---- CDNA5 async global->LDS staging (tracked by ASYNCcnt) ----
#if __has_builtin(__builtin_amdgcn_global_load_async_to_lds_b128)
__device__ __forceinline__ void async_g2l_16B(const float* g, float* l) {
  __builtin_amdgcn_global_load_async_to_lds_b128(
      (AS_GLOBAL v4i*)(g), (AS_LDS v4i*)(l), 0, 0);
}
#else
__device__ __forceinline__ void async_g2l_16B(const float* g, float* l) {
  AS_LDS char* lp = (AS_LDS char*)(l);
  asm volatile("global_load_async_to_lds_b128 %0, %1, off"
               :: "v"(lp), "v"(g) : "memory");
}
#endif

#if __has_builtin(__builtin_amdgcn_s_wait_asynccnt)
#define WAIT_ASYNCCNT(n) __builtin_amdgcn_s_wait_asynccnt(n)
#else
#define WAIT_ASYNCCNT(n) asm volatile("s_wait_asynccnt %0" :: "i"(n) : "memory")
#endif

// One wave32 per sequence. All control logic is wave-uniform (scores are
// scalars), so there is no divergence; lane L owns floats [4L,4L+4) of each
// 128-float key/val vector. cache_k|cache_v live in-place in d_out (exact
// output layout); cache_s lives in LDS; stored[]/start[] live in registers.
// The k/v token stream is staged through an 8-deep async LDS ring so the
// serial per-token chain only sees LDS latency, not HBM latency.
__global__ __launch_bounds__(32) void cascade_scan_kernel(
    const float* __restrict__ kIn, const float* __restrict__ vIn,
    const float* __restrict__ sIn, float* __restrict__ out) {
  const int seq  = blockIdx.x;
  const int lane = threadIdx.x;

  __shared__ __align__(16) float ring[PDEPTH * ROW];  // 8 KB k|v staging ring
  __shared__ float cacheS[T_SLOTS];                   // per-slot scores (8 KB)

  #pragma unroll
  for (int i = 0; i < T_SLOTS / 32; ++i) cacheS[i * 32 + lane] = 0.0f;

  const float* kseq = kIn + (size_t)seq * K_TOK * HID;
  const float* vseq = vIn + (size_t)seq * K_TOK * HID;
  const float* sseq = sIn + (size_t)seq * K_TOK;
  float*       oseq = out + (size_t)seq * T_SLOTS * ROW;

  int stored[CASCADES] = {0, 0, 0, 0};
  int start [CASCADES] = {0, 0, 0, 0};

  // prime the pipeline: stage tokens 0..PDEPTH-2 (2 async ops per token)
  #pragma unroll
  for (int p = 0; p < PDEPTH - 1; ++p) {
    const size_t go = (size_t)p * HID + (size_t)lane * 4;
    async_g2l_16B(kseq + go, &ring[p * ROW + lane * 4]);
    async_g2l_16B(vseq + go, &ring[p * ROW + HID + lane * 4]);
  }

  for (int idx = 0; idx < K_TOK; ++idx) {
    // Issue the copy for token idx+PDEPTH-1 into ring slot (idx-1)%PDEPTH
    // (already consumed last iteration). In the tail we keep issuing dummy
    // copies of token K-1 into that dead slot so exactly 2 ops are issued
    // per iteration and the wait immediate stays constant: waiting for
    // ASYNCcnt <= 2*(PDEPTH-1) retires token idx's two ops (in-order).
    {
      int pf = idx + PDEPTH - 1;
      if (pf >= K_TOK) pf = K_TOK - 1;              // dummy (dead-slot) copy
      const int    ps = (idx + PDEPTH - 1) & (PDEPTH - 1);
      const size_t go = (size_t)pf * HID + (size_t)lane * 4;
      async_g2l_16B(kseq + go, &ring[ps * ROW + lane * 4]);
      async_g2l_16B(vseq + go, &ring[ps * ROW + HID + lane * 4]);
    }
    WAIT_ASYNCCNT(2 * (PDEPTH - 1));
    asm volatile("" ::: "memory");  // compiler fence: re-read staged LDS data

    if (idx < NUM_SINK) continue;  // sink tokens never reach the cascade cache

    const int cur = idx & (PDEPTH - 1);
    float4 ck = *reinterpret_cast<const float4*>(&ring[cur * ROW + lane * 4]);
    float4 cv = *reinterpret_cast<const float4*>(&ring[cur * ROW + HID + lane * 4]);
    float  cs = sseq[idx];

    const bool warm = (idx <= MAX_SEQ);   // warm: every cascade accepts
    bool active = true;

    #pragma unroll
    for (int i = 0; i < CASCADES; ++i) {
      const bool dci = warm || (((idx - NUM_SINK) & ((1 << i) - 1)) == 0);
      const int  sti = stored[i];
      const int  si  = start[i];
      const bool notFull = sti < WINDOW;
      const bool writeA = active && ((dci && notFull) || (!dci && sti == 0));
      const bool writeB = active && dci && !notFull;
      const bool slotD  = active && !dci && (sti > 0);

      int tl;
      if (writeB)      tl = si;
      else if (slotD) { tl = si + sti - 1; if (tl >= WINDOW) tl -= WINDOW; }
      else             tl = si + sti;
      const int t = tl + i * WINDOW;

      const float olds  = cacheS[t];
      const bool writeD = slotD && (cs >= olds);
      const bool w      = writeA || writeB || writeD;

      float* slot = oseq + (size_t)t * ROW + lane * 4;
      float4 ok = make_float4(0.f, 0.f, 0.f, 0.f);
      float4 ov = make_float4(0.f, 0.f, 0.f, 0.f);
      if (writeB) {  // read evicted record before overwriting it
        ok = *reinterpret_cast<const float4*>(slot);
        ov = *reinterpret_cast<const float4*>(slot + HID);
      }
      if (w) {
        *reinterpret_cast<float4*>(slot)       = ck;
        *reinterpret_cast<float4*>(slot + HID) = cv;
        if (lane == 0) cacheS[t] = cs;
      }
      stored[i] = sti + (writeA ? 1 : 0);
      if (writeB) {
        const int nsi = si + 1;
        start[i] = (nsi >= WINDOW) ? 0 : nsi;
        ck = ok; cv = ov; cs = olds;   // carry evicted token downward
      }
      active = writeB;
      if (!active) break;  // deeper cascades unaffected
    }

    // CDNA5 prefetch hint: next eviction target of cascade 0 (hot path)
    __builtin_prefetch(oseq + (size_t)start[0] * ROW + lane * 4, 0, 1);
  }
}

__global__ void zero_f4_kernel(float4* __restrict__ p, int n4) {
  int i = blockIdx.x * blockDim.x + threadIdx.x;
  const int stride = gridDim.x * blockDim.x;
  const float4 z = make_float4(0.f, 0.f, 0.f, 0.f);
  for (; i < n4; i += stride) p[i] = z;
}

extern "C" void kernel_launch(void* const* d_in, const int* in_sizes, int n_in,
                              void* d_out, int out_size, void* d_ws, size_t ws_size,
                              hipStream_t stream) {
  (void)n_in; (void)d_ws; (void)ws_size;
  const float* k = (const float*)d_in[0];   // [N,H,K,HID] f32
  const float* v = (const float*)d_in[1];   // [N,H,K,HID] f32
  const float* s = (const float*)d_in[2];   // [N,H,K]     f32
  float* out = (float*)d_out;               // [N,H,T,2*HID] f32

  // cache starts as zeros and untouched slots must stay zero
  const int n4 = out_size / 4;
  zero_f4_kernel<<<2048, 256, 0, stream>>>((float4*)out, n4);

  const int nseq = in_sizes[2] / K_TOK;     // N*H independent sequences
  cascade_scan_kernel<<<nseq, 32, 0, stream>>>(k, v, s, out);
}